// GAT_SAGE_42271068127470
// MI455X (gfx1250) — compile-verified
//
#include <hip/hip_runtime.h>
#include <hip/hip_bf16.h>
#include <cstdint>
#include <cstddef>

// Problem constants (match reference)
constexpr int NND  = 50000;     // nodes
constexpr int NE   = 1600000;   // edges (before self-loops)
constexpr float NEG_SLOPE = 0.2f;
constexpr float BN_EPS    = 1e-5f;

typedef __attribute__((ext_vector_type(2))) float v2f;
typedef __attribute__((ext_vector_type(8))) float v8f;

__device__ __forceinline__ float lrelu(float x) { return x > 0.f ? x : NEG_SLOPE * x; }

// Order-preserving float -> u32 encoding for atomic max (handles negatives)
__device__ __forceinline__ unsigned enc_f(float v) {
    unsigned b = __float_as_uint(v);
    return b ^ ((b >> 31) ? 0xFFFFFFFFu : 0x80000000u);
}
__device__ __forceinline__ float dec_f(unsigned u) {
    return __uint_as_float(u ^ ((u >> 31) ? 0x80000000u : 0xFFFFFFFFu));
}

// ---------------------------------------------------------------------------
// fp32 WMMA GEMM: C[M x NC] = A[M x K] @ B[K x NC] (+C) (+bias) (relu) (BN)
// One wave computes a 16-row strip across the full NC width using
// V_WMMA_F32_16X16X4_F32. M must be a multiple of 16 (50000 = 3125*16).
// ---------------------------------------------------------------------------
template<int K, int NC, bool LOADC, bool BIAS, bool RELU, bool BN>
__global__ void __launch_bounds__(256)
gemm_f32_wmma(const float* __restrict__ A, const float* __restrict__ B,
              float* __restrict__ C, const float* __restrict__ bias,
              const float* __restrict__ bng, const float* __restrict__ bnb,
              const float* __restrict__ bnm, const float* __restrict__ bnv,
              int M)
{
    constexpr int JB = NC / 16;
    const int strip = blockIdx.x * (blockDim.x >> 5) + (threadIdx.x >> 5);
    if (strip * 16 >= M) return;                 // uniform per wave: EXEC stays all-1s
    const int lane = threadIdx.x & 31;
    const int half = lane >> 4;                  // 0: K pair {k,k+1}, 1: {k+2,k+3}
    const int r    = lane & 15;
    const int row0 = strip * 16;

    v8f acc[JB];
#pragma unroll
    for (int j = 0; j < JB; ++j) {
        if (LOADC) {
#pragma unroll
            for (int i = 0; i < 8; ++i)
                acc[j][i] = C[(size_t)(row0 + i + half * 8) * NC + j * 16 + r];
        } else {
#pragma unroll
            for (int i = 0; i < 8; ++i) acc[j][i] = 0.f;
        }
    }

    // A fragment (16x4 f32): lanes 0-15 hold K={k,k+1}, lanes 16-31 hold K={k+2,k+3}
    const float* Ar = A + (size_t)(row0 + r) * K + half * 2;
    for (int k = 0; k < K; k += 4) {
        v2f a;
        a.x = Ar[k];
        a.y = Ar[k + 1];
#pragma unroll
        for (int j = 0; j < JB; ++j) {
            // B fragment (4x16 f32): VGPR0 = row k (lanes 0-15) / k+2 (16-31), VGPR1 = k+1 / k+3
            v2f b;
            const int col = j * 16 + r;
            b.x = B[(size_t)(k + half * 2) * NC + col];
            b.y = B[(size_t)(k + half * 2 + 1) * NC + col];
            acc[j] = __builtin_amdgcn_wmma_f32_16x16x4_f32(
                false, a, false, b, (short)0, acc[j], false, false);
        }
    }

#pragma unroll
    for (int j = 0; j < JB; ++j) {
        const int col = j * 16 + r;
        float bsv = 0.f, sc = 1.f, mm = 0.f, bb = 0.f;
        if (BIAS) bsv = bias[col];
        if (BN) { sc = bng[col] * rsqrtf(bnv[col] + BN_EPS); mm = bnm[col]; bb = bnb[col]; }
#pragma unroll
        for (int i = 0; i < 8; ++i) {
            float v = acc[j][i];
            if (BIAS) v += bsv;
            if (RELU) v = fmaxf(v, 0.f);
            if (BN)   v = (v - mm) * sc + bb;
            C[(size_t)(row0 + i + half * 8) * NC + col] = v;
        }
    }
}

// ---------------------------------------------------------------------------
// Per-node attention logits: al_src[n,h] = <h[n,h,:], a_src[h,:]>, same for dst
// ---------------------------------------------------------------------------
__global__ void gat_logits_kernel(const float* __restrict__ h,
                                  const float* __restrict__ a_src,
                                  const float* __restrict__ a_dst,
                                  float* __restrict__ als, float* __restrict__ ald, int n)
{
    int t = blockIdx.x * blockDim.x + threadIdx.x;
    if (t >= n * 2) return;
    int node = t >> 1, hd = t & 1;
    const float* hr = h + (size_t)node * 128 + hd * 64;
    const float* as = a_src + hd * 64;
    const float* ad = a_dst + hd * 64;
    float s0 = 0.f, s1 = 0.f;
#pragma unroll
    for (int c = 0; c < 64; c += 4) {
        float4 hv = *(const float4*)(hr + c);
        float4 av = *(const float4*)(as + c);
        float4 dv = *(const float4*)(ad + c);
        s0 += hv.x * av.x + hv.y * av.y + hv.z * av.z + hv.w * av.w;
        s1 += hv.x * dv.x + hv.y * dv.y + hv.z * dv.z + hv.w * dv.w;
    }
    als[t] = s0;
    ald[t] = s1;
}

__global__ void zero_kernel(float* __restrict__ p, int n)
{
    int i = blockIdx.x * blockDim.x + threadIdx.x;
    if (i < n) p[i] = 0.f;
}

// Pass 1: segment max of leaky_relu(al_src[s]+al_dst[d]) over dst (incl. self-loops)
__global__ void gat_edge_max_kernel(const int* __restrict__ src, const int* __restrict__ dst,
                                    int ne, int nn,
                                    const float* __restrict__ als, const float* __restrict__ ald,
                                    unsigned* __restrict__ emax)
{
    int e = blockIdx.x * blockDim.x + threadIdx.x;
    if (e >= ne + nn) return;
    int s, d;
    if (e < ne) { s = src[e]; d = dst[e]; } else { s = d = e - ne; }
#pragma unroll
    for (int hd = 0; hd < 2; ++hd) {
        float v = lrelu(als[s * 2 + hd] + ald[d * 2 + hd]);
        atomicMax(&emax[d * 2 + hd], enc_f(v));
    }
}

// Pass 2: agg[d] += exp(e - max)*h[s], denom[d] += exp(e - max). 32 lanes/edge, float4 each.
__global__ void gat_edge_accum_kernel(const int* __restrict__ src, const int* __restrict__ dst,
                                      int ne, int nn,
                                      const float* __restrict__ als, const float* __restrict__ ald,
                                      const unsigned* __restrict__ emax,
                                      const float* __restrict__ h,
                                      float* __restrict__ agg, float* __restrict__ denom)
{
    int t = blockIdx.x * blockDim.x + threadIdx.x;
    int e = t >> 5, c = t & 31;
    if (e >= ne + nn) return;
    int s, d;
    if (e < ne) { s = src[e]; d = dst[e]; } else { s = d = e - ne; }
    const int hd = c >> 4;   // features [c*4, c*4+4) -> head c/16
    float v = lrelu(als[s * 2 + hd] + ald[d * 2 + hd]);
    float w = __expf(v - dec_f(emax[d * 2 + hd]));
    if ((c & 15) == 0) atomicAdd(&denom[d * 2 + hd], w);
    float4 hv = *(const float4*)(h + (size_t)s * 128 + c * 4);
    float* ap = agg + (size_t)d * 128 + c * 4;
    atomicAdd(ap + 0, w * hv.x);
    atomicAdd(ap + 1, w * hv.y);
    atomicAdd(ap + 2, w * hv.z);
    atomicAdd(ap + 3, w * hv.w);
}

// out = agg/denom + bias (+relu). Self-loops guarantee denom > 0.
template<bool RELU>
__global__ void gat_finalize_kernel(const float* __restrict__ agg, const float* __restrict__ denom,
                                    const float* __restrict__ bias, float* __restrict__ out, int n)
{
    int i = blockIdx.x * blockDim.x + threadIdx.x;
    if (i >= n * 128) return;
    int node = i >> 7, c = i & 127, hd = c >> 6;
    float v = agg[i] / denom[node * 2 + hd] + bias[c];
    if (RELU) v = fmaxf(v, 0.f);
    out[i] = v;
}

__global__ void sage_count_kernel(const int* __restrict__ dst, float* __restrict__ cnt, int ne)
{
    int e = blockIdx.x * blockDim.x + threadIdx.x;
    if (e < ne) atomicAdd(&cnt[dst[e]], 1.0f);
}

template<int F>
__global__ void sage_sum_kernel(const int* __restrict__ src, const int* __restrict__ dst,
                                const float* __restrict__ xf, float* __restrict__ agg, int ne)
{
    constexpr int CH = F / 4;
    int t = blockIdx.x * blockDim.x + threadIdx.x;
    int e = t / CH, c = t % CH;
    if (e >= ne) return;
    int s = src[e], d = dst[e];
    float4 v = *(const float4*)(xf + (size_t)s * F + c * 4);
    float* ap = agg + (size_t)d * F + c * 4;
    atomicAdd(ap + 0, v.x);
    atomicAdd(ap + 1, v.y);
    atomicAdd(ap + 2, v.z);
    atomicAdd(ap + 3, v.w);
}

template<int F>
__global__ void sage_mean_kernel(const float* __restrict__ agg, const float* __restrict__ cnt,
                                 float* __restrict__ mean, int n)
{
    int i = blockIdx.x * blockDim.x + threadIdx.x;
    if (i >= n * F) return;
    int node = i / F;
    mean[i] = agg[i] / fmaxf(cnt[node], 1.0f);
}

// ---------------------------------------------------------------------------
extern "C" void kernel_launch(void* const* d_in, const int* in_sizes, int n_in,
                              void* d_out, int out_size, void* d_ws, size_t ws_size,
                              hipStream_t stream)
{
    const float* x    = (const float*)d_in[0];
    const int*   ei   = (const int*)d_in[1];
    const float* W1   = (const float*)d_in[2];
    const float* as1  = (const float*)d_in[3];
    const float* ad1  = (const float*)d_in[4];
    const float* b1   = (const float*)d_in[5];
    const float* W2   = (const float*)d_in[6];
    const float* as2  = (const float*)d_in[7];
    const float* ad2  = (const float*)d_in[8];
    const float* b2   = (const float*)d_in[9];
    const float* s1wl = (const float*)d_in[10];
    const float* s1bl = (const float*)d_in[11];
    const float* s1wr = (const float*)d_in[12];
    const float* s2wl = (const float*)d_in[13];
    const float* s2bl = (const float*)d_in[14];
    const float* s2wr = (const float*)d_in[15];
    const float* bng  = (const float*)d_in[16];
    const float* bnb  = (const float*)d_in[17];
    const float* bnm  = (const float*)d_in[18];
    const float* bnv  = (const float*)d_in[19];

    const int* src = ei;          // edge_index[0]
    const int* dst = ei + NE;     // edge_index[1]
    float* out = (float*)d_out;

    // Workspace layout (floats)
    float* ws      = (float*)d_ws;
    float* hbuf    = ws;                       // N*128  (GAT h; later aliased as meanbuf)
    float* bufA    = hbuf + (size_t)NND * 128; // N*128  (gat1 out; later s1out + mean2)
    float* bufB    = bufA + (size_t)NND * 128; // N*128  (gat2 out = SAGE input)
    float* aggbuf  = bufB + (size_t)NND * 128; // N*128  (segment sums)
    float* al_s    = aggbuf + (size_t)NND * 128; // N*2
    float* al_d    = al_s + NND * 2;             // N*2
    float* denom   = al_d + NND * 2;             // N*2
    unsigned* emax = (unsigned*)(denom + NND * 2); // N*2 (encoded floats; 0 == -inf)
    float* cnt     = (float*)(emax + NND * 2);   // N
    float* meanbuf = hbuf;                 // alias (hbuf dead after GAT2 accumulate)
    float* s1out   = bufA;                 // alias (bufA dead after GAT2 GEMM)
    float* mean2   = bufA + (size_t)NND * 64;

    constexpr int TB = 256;
    auto blk = [](long long n) { return (int)((n + TB - 1) / TB); };
    const int gemm_grid = (NND / 16 + 7) / 8;   // 3125 strips, 8 waves/block
    const int Etot = NE + NND;

    // ---- GAT layer 1 ----
    gemm_f32_wmma<256, 128, false, false, false, false>
        <<<gemm_grid, TB, 0, stream>>>(x, W1, hbuf, nullptr, nullptr, nullptr, nullptr, nullptr, NND);
    gat_logits_kernel<<<blk(NND * 2), TB, 0, stream>>>(hbuf, as1, ad1, al_s, al_d, NND);
    zero_kernel<<<blk((long long)NND * 128), TB, 0, stream>>>(aggbuf, NND * 128);
    zero_kernel<<<blk(NND * 2), TB, 0, stream>>>(denom, NND * 2);
    zero_kernel<<<blk(NND * 2), TB, 0, stream>>>((float*)emax, NND * 2);
    gat_edge_max_kernel<<<blk(Etot), TB, 0, stream>>>(src, dst, NE, NND, al_s, al_d, emax);
    gat_edge_accum_kernel<<<blk((long long)Etot * 32), TB, 0, stream>>>(
        src, dst, NE, NND, al_s, al_d, emax, hbuf, aggbuf, denom);
    gat_finalize_kernel<true><<<blk((long long)NND * 128), TB, 0, stream>>>(aggbuf, denom, b1, bufA, NND);

    // ---- GAT layer 2 ----
    gemm_f32_wmma<128, 128, false, false, false, false>
        <<<gemm_grid, TB, 0, stream>>>(bufA, W2, hbuf, nullptr, nullptr, nullptr, nullptr, nullptr, NND);
    gat_logits_kernel<<<blk(NND * 2), TB, 0, stream>>>(hbuf, as2, ad2, al_s, al_d, NND);
    zero_kernel<<<blk((long long)NND * 128), TB, 0, stream>>>(aggbuf, NND * 128);
    zero_kernel<<<blk(NND * 2), TB, 0, stream>>>(denom, NND * 2);
    zero_kernel<<<blk(NND * 2), TB, 0, stream>>>((float*)emax, NND * 2);
    gat_edge_max_kernel<<<blk(Etot), TB, 0, stream>>>(src, dst, NE, NND, al_s, al_d, emax);
    gat_edge_accum_kernel<<<blk((long long)Etot * 32), TB, 0, stream>>>(
        src, dst, NE, NND, al_s, al_d, emax, hbuf, aggbuf, denom);
    gat_finalize_kernel<false><<<blk((long long)NND * 128), TB, 0, stream>>>(aggbuf, denom, b2, bufB, NND);

    // ---- SAGE layer 1 (input bufB, 128 feats) ----
    zero_kernel<<<blk(NND), TB, 0, stream>>>(cnt, NND);
    zero_kernel<<<blk((long long)NND * 128), TB, 0, stream>>>(aggbuf, NND * 128);
    sage_count_kernel<<<blk(NE), TB, 0, stream>>>(dst, cnt, NE);
    sage_sum_kernel<128><<<blk((long long)NE * 32), TB, 0, stream>>>(src, dst, bufB, aggbuf, NE);
    sage_mean_kernel<128><<<blk((long long)NND * 128), TB, 0, stream>>>(aggbuf, cnt, meanbuf, NND);
    gemm_f32_wmma<128, 64, false, false, false, false>
        <<<gemm_grid, TB, 0, stream>>>(bufB, s1wr, s1out, nullptr, nullptr, nullptr, nullptr, nullptr, NND);
    gemm_f32_wmma<128, 64, true, true, true, false>
        <<<gemm_grid, TB, 0, stream>>>(meanbuf, s1wl, s1out, s1bl, nullptr, nullptr, nullptr, nullptr, NND);

    // ---- SAGE layer 2 (input s1out, 64 feats) + BatchNorm ----
    zero_kernel<<<blk((long long)NND * 64), TB, 0, stream>>>(aggbuf, NND * 64);
    sage_sum_kernel<64><<<blk((long long)NE * 16), TB, 0, stream>>>(src, dst, s1out, aggbuf, NE);
    sage_mean_kernel<64><<<blk((long long)NND * 64), TB, 0, stream>>>(aggbuf, cnt, mean2, NND);
    gemm_f32_wmma<64, 64, false, false, false, false>
        <<<gemm_grid, TB, 0, stream>>>(s1out, s2wr, out, nullptr, nullptr, nullptr, nullptr, nullptr, NND);
    gemm_f32_wmma<64, 64, true, true, true, true>
        <<<gemm_grid, TB, 0, stream>>>(mean2, s2wl, out, s2bl, bng, bnb, bnm, bnv, NND);
}